// MS_22187801051864
// MI455X (gfx1250) — compile-verified
//
#include <hip/hip_runtime.h>
#include <math.h>

// Problem constants (from reference)
#define BATCH   2
#define DIM     64
#define HW      9216          // 96*96
#define ITERS   4
#define BWD     0.3f
#define STEPSZ  0.5f

// Tiling
#define MT 64                 // output columns per block
#define NT 32                 // rows (n) processed per inner step
#define NTILES (HW / NT)      // 288
#define XM_PAD 4              // floats (16B) -> odd 4B-bank stride, conflict free
#define XN_PAD 4
#define K_PAD  4

typedef float v2f __attribute__((ext_vector_type(2)));
typedef float v8f __attribute__((ext_vector_type(8)));
typedef unsigned int u32;
typedef u32 u32x4 __attribute__((ext_vector_type(4)));
typedef int  i32x4 __attribute__((ext_vector_type(4)));
typedef int  i32x8 __attribute__((ext_vector_type(8)));

#if __has_builtin(__builtin_amdgcn_tensor_load_to_lds) && \
    __has_builtin(__builtin_amdgcn_s_wait_tensorcnt)
#define USE_TDM 1
#else
#define USE_TDM 0
#endif

// ---------------------------------------------------------------------------
// Copy x_in -> output slot 0 (out[b][0][:,:] = x_in[b])
// ---------------------------------------------------------------------------
__global__ void MS_copy_x0(const float* __restrict__ xin, float* __restrict__ out) {
    const size_t DN = (size_t)DIM * HW;
    size_t i = (size_t)blockIdx.x * blockDim.x + threadIdx.x;
    if (i < (size_t)BATCH * DN) {
        size_t b = i / DN;
        size_t r = i % DN;
        out[b * (size_t)(ITERS + 1) * DN + r] = xin[i];
    }
}

#if USE_TDM
// ---------------------------------------------------------------------------
// TDM 2D tile load: DIM rows (dim1) x tile_n elems (dim0) of f32, row stride
// HW elements in global memory, padded rows in LDS.
//   pad_interval code: 4 -> pad after every 32 dwords, 5 -> after every 64
//   pad_amount   code: 3 -> skip 4 dwords (16B) of LDS after each interval
// Descriptor bitfields per CDNA5 ISA 8.3/8.4 (D# group0/group1).
// This toolchain exposes the 6-arg builtin:
//   (u32x4 g0, i32x8 g1, i32x4 g2, i32x4 g3, i32x8 extra, i32 cpol)
// ---------------------------------------------------------------------------
__device__ __forceinline__ void tdm_load_2d(u32 lds_off, const float* gptr,
                                            u32 tile_n, u32 tile_d,
                                            u32 pad_interval, u32 pad_amount) {
    unsigned long long ga = (unsigned long long)(uintptr_t)gptr;
    u32x4 g0;
    g0.x = 1u;                                        // count=1, user mode
    g0.y = lds_off;                                   // lds_addr (bytes)
    g0.z = (u32)(ga & 0xFFFFFFFFu);                   // global_addr[31:0]
    g0.w = (u32)((ga >> 32) & 0x01FFFFFFu) | (2u << 30); // addr[56:32] | type=2

    i32x8 g1;
    g1[0] = (int)((2u << 16)                          // data_size = 4B
                | (1u << 20)                          // pad_enable
                | (pad_interval << 22)
                | (pad_amount << 25));
    g1[1] = (int)(((u32)HW & 0xFFFFu) << 16);         // tensor_dim0[15:0]
    g1[2] = (int)((((u32)HW >> 16) & 0xFFFFu)         // tensor_dim0[31:16]
                | (((u32)DIM & 0xFFFFu) << 16));      // tensor_dim1[15:0]
    g1[3] = (int)((tile_n & 0xFFFFu) << 16);          // tile_dim0
    g1[4] = (int)(tile_d & 0xFFFFu);                  // tile_dim1 (tile_dim2=0)
    g1[5] = (int)HW;                                  // tensor_dim0_stride[31:0]
    g1[6] = 0;                                        // stride hi / dim1_stride lo
    g1[7] = 0;

    i32x4 z4 = {0, 0, 0, 0};                          // groups 2/3 unused (2D)
    i32x8 z8 = {0, 0, 0, 0, 0, 0, 0, 0};
    __builtin_amdgcn_tensor_load_to_lds(g0, g1, z4, z4, z8, 0);
}
#endif

// ---------------------------------------------------------------------------
// One mean-shift iteration, fused flash-attention style.
// Block: batch b = blockIdx.y, output columns [m0, m0+MT).
// ---------------------------------------------------------------------------
__global__ __launch_bounds__(256) void MS_msiter(const float* __restrict__ xprev,
                                                 float* __restrict__ xnew) {
    const size_t DN     = (size_t)DIM * HW;
    const size_t SLOT_B = (size_t)(ITERS + 1) * DN;   // per-batch stride in out buf
    const int    b      = blockIdx.y;
    const int    m0     = blockIdx.x * MT;

    const float* __restrict__ Xb = xprev + (size_t)b * SLOT_B;
    float*       __restrict__ Yb = xnew  + (size_t)b * SLOT_B;

    __shared__ float xm[DIM][MT + XM_PAD];       // X[:, m-tile], block-resident
    __shared__ float xn[2][DIM][NT + XN_PAD];    // X[:, n-tile], double-buffered
    __shared__ float kt[NT][MT + K_PAD];         // exp(bw * S) tile
    __shared__ float part[4][MT];                // per-lane colsum partials
    __shared__ float cs[MT];                     // final column sums

    const int tid  = threadIdx.x;
    const int w    = tid >> 5;                   // wave 0..7
    const int lane = tid & 31;
    const int lh   = lane & 15;
    const bool hi  = lane >= 16;
    const int r    = w >> 2;                     // wave row 0..1 (n-subtile, GEMM1)
    const int cg   = w & 3;                      // wave col 0..3 (m-subtile, both)

    // ---- Prologue: stage xm tile + first xn tile ---------------------------
#if USE_TDM
    if (w == 0) {
        tdm_load_2d((u32)(uintptr_t)&xm[0][0],    Xb + m0, MT, DIM, 5u, 3u);
        tdm_load_2d((u32)(uintptr_t)&xn[0][0][0], Xb,      NT, DIM, 4u, 3u);
        __builtin_amdgcn_s_wait_tensorcnt(0);
    }
#else
    for (int idx4 = tid; idx4 < DIM * MT / 4; idx4 += 256) {
        int d = idx4 >> 4, m4 = (idx4 & 15) * 4;
        float4 v = *(const float4*)&Xb[(size_t)d * HW + m0 + m4];
        *(float4*)&xm[d][m4] = v;
    }
    for (int idx4 = tid; idx4 < DIM * NT / 4; idx4 += 256) {
        int d = idx4 >> 3, n4 = (idx4 & 7) * 4;
        float4 v = *(const float4*)&Xb[(size_t)d * HW + n4];
        *(float4*)&xn[0][d][n4] = v;
    }
#endif
    __syncthreads();

    float csum = 0.0f;                           // thread tid<64 owns col m0+tid
    v8f acc[2] = {};                             // xk accumulators (d-subtiles)

    for (int nb = 0; nb < NTILES; ++nb) {
        const int cur = nb & 1;
        const float (*__restrict__ XN)[NT + XN_PAD] = xn[cur];

        // Kick DMA for next tile while this one is computed. The previous
        // reader of buffer cur^1 finished before the trailing barrier of the
        // previous step, so the overwrite is safe.
#if USE_TDM
        if (w == 0 && nb + 1 < NTILES) {
            tdm_load_2d((u32)(uintptr_t)&xn[cur ^ 1][0][0],
                        Xb + (size_t)(nb + 1) * NT, NT, DIM, 4u, 3u);
        }
#else
        if (nb + 1 < NTILES) {
            const size_t n1 = (size_t)(nb + 1) * NT;
            for (int idx4 = tid; idx4 < DIM * NT / 4; idx4 += 256) {
                int d = idx4 >> 3, n4 = (idx4 & 7) * 4;
                float4 v = *(const float4*)&Xb[(size_t)d * HW + n1 + n4];
                *(float4*)&xn[cur ^ 1][d][n4] = v;
            }
        }
#endif

        // ---- GEMM1: S(32x64) = Xn^T * Xm, contraction over d = 64 ----------
        v8f s = {};
        #pragma unroll
        for (int kk = 0; kk < DIM / 4; ++kk) {
            const int kb = kk * 4 + (hi ? 2 : 0);
            v2f af, bf;
            af.x = XN[kb + 0][r * 16 + lh];      // A[i][k] = x[d=k][n]
            af.y = XN[kb + 1][r * 16 + lh];
            bf.x = xm[kb + 0][cg * 16 + lh];     // B[k][j] = x[d=k][m]
            bf.y = xm[kb + 1][cg * 16 + lh];
            s = __builtin_amdgcn_wmma_f32_16x16x4_f32(
                    false, af, false, bf, (short)0, s, false, false);
        }

        // ---- K = exp(bw*S): write tile + in-register column partial --------
        float psum = 0.0f;
        #pragma unroll
        for (int v = 0; v < 8; ++v) {
            float kv = __expf(BWD * s[v]);
            int row = r * 16 + (hi ? v + 8 : v);
            kt[row][cg * 16 + lh] = kv;
            psum += kv;
        }
        part[r * 2 + (hi ? 1 : 0)][cg * 16 + lh] = psum;
        __syncthreads();

        // ---- colsum accumulate (deterministic fixed order) -----------------
        if (tid < MT)
            csum += (part[0][tid] + part[1][tid]) + (part[2][tid] + part[3][tid]);

        // ---- GEMM2: xk(64x64) += Xn(64x32) * K(32x64) ----------------------
        #pragma unroll
        for (int half = 0; half < 2; ++half) {
            const int dbase = (r * 2 + half) * 16;
            #pragma unroll
            for (int kk = 0; kk < NT / 4; ++kk) {
                const int kb = kk * 4 + (hi ? 2 : 0);
                v2f af, bf;
                af.x = XN[dbase + lh][kb + 0];   // A[i][k] = x[d][n0+k]
                af.y = XN[dbase + lh][kb + 1];
                bf.x = kt[kb + 0][cg * 16 + lh]; // B[k][j] = K[n0+k][m]
                bf.y = kt[kb + 1][cg * 16 + lh];
                acc[half] = __builtin_amdgcn_wmma_f32_16x16x4_f32(
                                false, af, false, bf, (short)0, acc[half], false, false);
            }
        }

#if USE_TDM
        if (w == 0 && nb + 1 < NTILES)
            __builtin_amdgcn_s_wait_tensorcnt(0);   // next tile landed in LDS
#endif
        __syncthreads();   // protect xn (next buf) / kt before next step
    }

    // ---- Finalize: x_new = 0.5 * xk/colsum + 0.5 * x_old -------------------
    if (tid < MT) cs[tid] = csum;
    __syncthreads();

    #pragma unroll
    for (int half = 0; half < 2; ++half) {
        const int dbase = (r * 2 + half) * 16;
        #pragma unroll
        for (int v = 0; v < 8; ++v) {
            int d  = dbase + (hi ? v + 8 : v);
            int ml = cg * 16 + lh;
            float val = STEPSZ * (acc[half][v] / cs[ml])
                      + (1.0f - STEPSZ) * xm[d][ml];
            Yb[(size_t)d * HW + (m0 + ml)] = val;
        }
    }
}

// ---------------------------------------------------------------------------
extern "C" void kernel_launch(void* const* d_in, const int* in_sizes, int n_in,
                              void* d_out, int out_size, void* d_ws, size_t ws_size,
                              hipStream_t stream) {
    (void)in_sizes; (void)n_in; (void)out_size; (void)d_ws; (void)ws_size;
    const float* xin = (const float*)d_in[0];
    float* out = (float*)d_out;
    const size_t DN = (size_t)DIM * HW;

    // Slot 0 = x0
    {
        size_t total = (size_t)BATCH * DN;
        int blocks = (int)((total + 255) / 256);
        MS_copy_x0<<<blocks, 256, 0, stream>>>(xin, out);
    }

    // Iterations: slot i reads slot i-1 (out buffer doubles as ping-pong state)
    dim3 grid(HW / MT, BATCH);
    for (int it = 1; it <= ITERS; ++it) {
        const float* xprev = out + (size_t)(it - 1) * DN;
        float*       xnext = out + (size_t)it * DN;
        MS_msiter<<<grid, 256, 0, stream>>>(xprev, xnext);
    }
}